// LASTAggregator_70214125355180
// MI455X (gfx1250) — compile-verified
//
#include <hip/hip_runtime.h>
#include <hip/hip_bf16.h>

// Problem constants (match reference)
#define Bsz   64
#define Npat  1024
#define Dh    768
#define KTOP  8
#define CP_STRIDE 770   // 768 + 2 pad -> LDS bank-conflict-free float2 reads

typedef __attribute__((ext_vector_type(2))) float v2f;
typedef __attribute__((ext_vector_type(8))) float v8f;

// ---------------------------------------------------------------------------
// Kernel 1: build the circulant impulse response h[p] = (1/D) * sum_k g[k]cos(2pi k p / D)
// with g[k] = kernel[(k + D/2) % D], kernel[j] = exp(-0.5*((j-(D/2-1))/sqrt(D))^2)
// (kernel.max()==1 exactly, so no renormalization). Replicates the reference's
// fftshift/ifftshift alignment including its off-center gaussian.
// ---------------------------------------------------------------------------
__global__ __launch_bounds__(Dh) void build_h_kernel(float* __restrict__ h) {
  __shared__ float g[Dh];
  const int t = threadIdx.x;
  const float inv_sigma2 = 1.0f / (float)Dh;   // sigma^2 = D
  int j = (t + Dh / 2) % Dh;
  float pos = (float)(j - (Dh / 2 - 1));
  g[t] = expf(-0.5f * pos * pos * inv_sigma2);
  __syncthreads();

  float sum = 0.0f;
  const float w = 6.28318530717958647692f / (float)Dh;
  for (int k = 0; k < Dh; ++k) {
    int m = (k * t) % Dh;                       // exact angle reduction
    sum += g[k] * cosf(w * (float)m);
  }
  h[t] = sum * (1.0f / (float)Dh);
}

// sorted-descending top-8 insert; total order (value desc, index asc) matches
// jax.lax.top_k stability, and makes merge order irrelevant. Early-reject is a
// branch (cheap, almost always taken); the insertion network is branchless
// selects so it lowers to v_cmp + v_cndmask instead of exec-mask ladders.
__device__ __forceinline__ void topk_insert(float cv, int ci, float tv[KTOP], int ti[KTOP]) {
  if (cv < tv[KTOP - 1] || (cv == tv[KTOP - 1] && ci > ti[KTOP - 1])) return;
#pragma unroll
  for (int j = 0; j < KTOP; ++j) {
    bool better = (cv > tv[j]) || (cv == tv[j] && ci < ti[j]);
    float ov = tv[j]; int oi = ti[j];
    tv[j] = better ? cv : ov;
    ti[j] = better ? ci : oi;
    cv    = better ? ov : cv;
    ci    = better ? oi : ci;
  }
}

struct CandBuf {
  float vals[16][16][KTOP];   // [list][channel][slot]
  int   idx [16][16][KTOP];
};
union SMemU {
  float   cp[16 * CP_STRIDE]; // operator panel, 49280 B (live during GEMM)
  CandBuf cand;               // 16384 B merge buffer (live after GEMM)
};

// ---------------------------------------------------------------------------
// Kernel 2: fused  filtered = X @ C^T (f32 WMMA)  ->  scores  ->  per-channel
// top-8 over the 1024 patches -> pooled / votes / sel_idx.
// grid = (48, 64): one workgroup per (16-channel tile, batch). 8 waves, each
// wave owns 8 of the 64 patch-tiles (16 rows each).
// ---------------------------------------------------------------------------
__global__ __launch_bounds__(256) void last_main_kernel(
    const float* __restrict__ X, const float* __restrict__ h,
    float* __restrict__ pooled, int* __restrict__ votes, int* __restrict__ sel) {
  __shared__ SMemU smem;

  const int b    = blockIdx.y;
  const int d0   = blockIdx.x * 16;
  const int tid  = threadIdx.x;
  const int lane = tid & 31;
  const int wave = tid >> 5;
  const int col  = lane & 15;   // lane % 16 -> channel within tile / A-row
  const int half = lane >> 4;   // lane / 16 -> K split for A/B operands

  // Fill operator panel: cp[d][k] = C^T[k][d0+d] = h[(d0 + d - k) mod 768]
  for (int i = tid; i < 16 * Dh; i += 256) {
    int d = i / Dh;
    int k = i - d * Dh;
    int p = d0 + d - k;
    if (p < 0) p += Dh;
    smem.cp[d * CP_STRIDE + k] = h[p];
  }
  __syncthreads();

  const float* xb = X + (size_t)b * Npat * Dh;

  float tv[KTOP]; int ti[KTOP];
#pragma unroll
  for (int i = 0; i < KTOP; ++i) { tv[i] = -__builtin_inff(); ti[i] = 0x7fffffff; }

  const float* cpL = smem.cp + col * CP_STRIDE + 2 * half;   // B operand base

  for (int t = wave * 8; t < wave * 8 + 8; ++t) {
    const int n0 = t * 16;
    // Two independent accumulator chains -> two independent WMMA dependency
    // streams per wave (hides XDL latency even at modest occupancy).
    v8f acc0 = {0.f, 0.f, 0.f, 0.f, 0.f, 0.f, 0.f, 0.f};
    v8f acc1 = {0.f, 0.f, 0.f, 0.f, 0.f, 0.f, 0.f, 0.f};
    // A layout (f32 16x4): lane%16 = M (patch row), VGPR pair = consecutive K,
    // lanes 16-31 handle K+2 -> each lane reads an 8B-aligned float2.
    const float* aL = xb + (size_t)(n0 + col) * Dh + 2 * half;
#pragma unroll 2
    for (int k0 = 0; k0 < Dh; k0 += 8) {
      v2f a0 = *(const v2f*)(aL + k0);
      v2f b0 = *(const v2f*)(cpL + k0);
      v2f a1 = *(const v2f*)(aL + k0 + 4);
      v2f b1 = *(const v2f*)(cpL + k0 + 4);
      acc0 = __builtin_amdgcn_wmma_f32_16x16x4_f32(
          false, a0, false, b0, (short)0, acc0, false, false);
      acc1 = __builtin_amdgcn_wmma_f32_16x16x4_f32(
          false, a1, false, b1, (short)0, acc1, false, false);
    }
    // Epilogue: C/D layout -> lane holds rows (n0 + v + 8*half) of channel col.
#pragma unroll
    for (int v = 0; v < 8; ++v) {
      int n = n0 + v + 8 * half;
      float x = xb[(size_t)n * Dh + d0 + col];
      float f = acc0[v] + acc1[v];
      float s = x / fmaxf(fabsf(f - x), 1e-6f);
      topk_insert(s, n, tv, ti);
    }
  }

  __syncthreads();   // all waves done reading cp -> safe to alias
  {
    int list = wave * 2 + half;
#pragma unroll
    for (int i = 0; i < KTOP; ++i) {
      smem.cand.vals[list][col][i] = tv[i];
      smem.cand.idx [list][col][i] = ti[i];
    }
  }
  __syncthreads();

  // Final merge: 16 threads, one per channel; union of 16 partial top-8 lists
  // necessarily contains the global top-8.
  if (tid < 16) {
    const int c = tid;
    float mv[KTOP]; int mi[KTOP];
#pragma unroll
    for (int i = 0; i < KTOP; ++i) { mv[i] = -__builtin_inff(); mi[i] = 0x7fffffff; }
    for (int list = 0; list < 16; ++list) {
#pragma unroll
      for (int s = 0; s < KTOP; ++s) {
        topk_insert(smem.cand.vals[list][c][s], smem.cand.idx[list][c][s], mv, mi);
      }
    }
    const int dcol = d0 + c;
    float sum = 0.0f;
#pragma unroll
    for (int i = 0; i < KTOP; ++i) {
      int n = mi[i];
      sel[((size_t)b * KTOP + i) * Dh + dcol] = n;
      sum += xb[(size_t)n * Dh + dcol];
      atomicAdd(&votes[b * Npat + n], 1);
    }
    pooled[(size_t)b * Dh + dcol] = sum * (1.0f / (float)KTOP);
  }
}

// ---------------------------------------------------------------------------
extern "C" void kernel_launch(void* const* d_in, const int* in_sizes, int n_in,
                              void* d_out, int out_size, void* d_ws, size_t ws_size,
                              hipStream_t stream) {
  const float* X = (const float*)d_in[0];

  // d_out layout (return order, all 4-byte elements):
  //   pooled  [B, D]    float32
  //   votes   [B, N]    int32
  //   sel_idx [B, K, D] int32
  float* pooled = (float*)d_out;
  int*   votes  = (int*)((float*)d_out + (size_t)Bsz * Dh);
  int*   sel    = (int*)((float*)d_out + (size_t)Bsz * Dh + (size_t)Bsz * Npat);

  float* h = (float*)d_ws;   // 768 floats of scratch

  hipMemsetAsync(votes, 0, (size_t)Bsz * Npat * sizeof(int), stream);
  build_h_kernel<<<1, Dh, 0, stream>>>(h);

  dim3 grid(Dh / 16, Bsz);
  last_main_kernel<<<grid, 256, 0, stream>>>(X, h, pooled, votes, sel);
}